// RandomFilter_40767829574170
// MI455X (gfx1250) — compile-verified
//
#include <hip/hip_runtime.h>

// ---------------------------------------------------------------------------
// RandomFilter: per-batch 3x3 cross-correlation, SAME zero padding,
// NaN->1.0, clip to [0,1].  out[b,c,y,x] = sum_{i,j} w[b,i,j]*img[b,c,y+i-1,x+j-1]
// w[b,i,j] = 0.1*noise[b,i,j] + (i==2 && j==2)
//
// Memory-bound (AI ~2.25 FLOP/B -> ~8.6us floor at 23.3 TB/s). Strategy:
// async global->LDS tile staging (CDNA5 GLOBAL_LOAD_ASYNC_TO_LDS_B128 +
// s_wait_asynccnt), 9-tap stencil from LDS, non-temporal 128-bit streaming
// stores (output is write-once; keep L2 for the input halo re-reads).
// ---------------------------------------------------------------------------

#ifndef __has_builtin
#define __has_builtin(x) 0
#endif

#if __has_builtin(__builtin_amdgcn_global_load_async_to_lds_b128)
#define RF_ASYNC_LDS 1
#else
#define RF_ASYNC_LDS 0
#endif

// Matches the builtin's parameter type per hipcc diagnostic:
// 'int __attribute__((vector_size(4*sizeof(int)))) *'
typedef int   rf_v4i __attribute__((vector_size(16)));
typedef float rf_v4f __attribute__((vector_size(16)));

namespace {
constexpr int IMG_W   = 512;
constexpr int IMG_H   = 512;
constexpr int TILE_R  = 16;            // output rows per workgroup
constexpr int HALO_R  = TILE_R + 2;    // staged rows (1 halo above/below)
constexpr int NBATCH  = 32;
constexpr int NCHAN   = 3;
constexpr float SIGMA = 0.1f;
}

__device__ __forceinline__ float rf_finalize(float v) {
  v = (v != v) ? 1.0f : v;             // NaN -> 1.0
  return fminf(fmaxf(v, 0.0f), 1.0f);  // clip [0,1]
}

__global__ __launch_bounds__(256) void
RandomFilter_40767829574170_kernel(const float* __restrict__ img,
                                   const float* __restrict__ noise,
                                   float* __restrict__ out) {
  __shared__ float tile[HALO_R][IMG_W];   // 18 * 512 * 4 = 36,864 B
  __shared__ float wsm[9];

  const int tid = threadIdx.x;            // 0..255 (8 wave32s)
  const int b   = blockIdx.z;
  const int c   = blockIdx.y;
  const int y0  = blockIdx.x * TILE_R;

  const size_t plane = (size_t)(b * NCHAN + c) * (size_t)(IMG_H * IMG_W);
  const float* __restrict__ gplane = img + plane;

  // Per-batch 3x3 weights: w = SIGMA*noise + delta at (2,2) (flat index 8).
  if (tid < 9) {
    const float delta = (tid == 8) ? 1.0f : 0.0f;
    wsm[tid] = SIGMA * noise[b * 9 + tid] + delta;
  }

  // ---- Stage HALO_R x IMG_W floats into LDS with 128-bit async copies ----
  // 18 rows * 128 b128-chunks = 2304 chunks; 9 chunks per thread.
#pragma unroll
  for (int k = 0; k < 9; ++k) {
    const int idx = tid + k * 256;        // 0..2303
    const int lr  = idx >> 7;             // staged row 0..17
    const int cx  = (idx & 127) << 2;     // float column 0..508, step 4
    const int gy  = y0 - 1 + lr;          // global source row
    float* lp = &tile[lr][cx];
    if (gy >= 0 && gy < IMG_H) {
      const float* gp = gplane + (size_t)gy * IMG_W + cx;
#if RF_ASYNC_LDS
      __builtin_amdgcn_global_load_async_to_lds_b128(
          (rf_v4i*)gp, (rf_v4i*)lp, /*offset=*/0, /*cpol=*/0);
#else
      *(float4*)lp = *(const float4*)gp;  // fallback: vmem load + ds store
#endif
    } else {
      *(float4*)lp = make_float4(0.f, 0.f, 0.f, 0.f);  // zero padding rows
    }
  }

#if RF_ASYNC_LDS
#if __has_builtin(__builtin_amdgcn_s_wait_asynccnt)
  __builtin_amdgcn_s_wait_asynccnt(0);
#else
  asm volatile("s_wait_asynccnt 0" ::: "memory");
#endif
#endif
  __syncthreads();

  // ---- 9-tap stencil from LDS, 4 outputs per iteration, 8 iterations ----
  float* __restrict__ oplane = out + plane;
#pragma unroll
  for (int k = 0; k < 8; ++k) {
    const int idx = tid + k * 256;        // 0..2047
    const int r   = idx >> 7;             // output row in tile 0..15
    const int cx  = (idx & 127) << 2;     // output column 0..508, step 4
    float a0 = 0.f, a1 = 0.f, a2 = 0.f, a3 = 0.f;
#pragma unroll
    for (int i = 0; i < 3; ++i) {
      // output row y0+r reads input rows y0+r+i-1 == staged row r+i
      const float* row = &tile[r + i][0];
      const float xm = (cx > 0)          ? row[cx - 1] : 0.0f;  // left pad
      const float x0 = row[cx + 0];
      const float x1 = row[cx + 1];
      const float x2 = row[cx + 2];
      const float x3 = row[cx + 3];
      const float xp = (cx < IMG_W - 4)  ? row[cx + 4] : 0.0f;  // right pad
      const float w0 = wsm[i * 3 + 0];
      const float w1 = wsm[i * 3 + 1];
      const float w2 = wsm[i * 3 + 2];
      a0 = fmaf(w0, xm, fmaf(w1, x0, fmaf(w2, x1, a0)));
      a1 = fmaf(w0, x0, fmaf(w1, x1, fmaf(w2, x2, a1)));
      a2 = fmaf(w0, x1, fmaf(w1, x2, fmaf(w2, x3, a2)));
      a3 = fmaf(w0, x2, fmaf(w1, x3, fmaf(w2, xp, a3)));
    }
    rf_v4f o = { rf_finalize(a0), rf_finalize(a1),
                 rf_finalize(a2), rf_finalize(a3) };
    // Non-temporal 128-bit store: output is write-once, don't pollute L2.
    __builtin_nontemporal_store(
        o, (rf_v4f*)(oplane + (size_t)(y0 + r) * IMG_W + cx));
  }
}

extern "C" void kernel_launch(void* const* d_in, const int* in_sizes, int n_in,
                              void* d_out, int out_size, void* d_ws, size_t ws_size,
                              hipStream_t stream) {
  (void)in_sizes; (void)n_in; (void)out_size; (void)d_ws; (void)ws_size;
  const float* img   = (const float*)d_in[0];   // [32,3,512,512] f32
  const float* noise = (const float*)d_in[1];   // [32,3,3] f32
  float* out         = (float*)d_out;           // [32,3,512,512] f32

  dim3 grid(IMG_H / TILE_R, NCHAN, NBATCH);     // (32, 3, 32)
  dim3 block(256);
  RandomFilter_40767829574170_kernel<<<grid, block, 0, stream>>>(img, noise, out);
}